// QuantumRegressionGen_65481071398329
// MI455X (gfx1250) — compile-verified
//
#include <hip/hip_runtime.h>

#define NW     10
#define NOPS   30
#define BT     16          // batch elements per workgroup (LDS tile: 16 x 8KB state = 128KB)
#define TPB    256         // 8 waves of 32
#define NWAVES (TPB / 32)
#define NSTATE 1024        // 2^10 amplitudes

typedef __attribute__((ext_vector_type(2))) float v2f;
typedef __attribute__((ext_vector_type(8))) float v8f;

// ---------------------------------------------------------------------------
// Frozen 30-op random circuit (structure from the reference's rng(42) op list,
// generated offline).  kind: 0=rx, 1=ry, 2=rz, 3=cnot(a=control, b=target).
// Angles come from rand_params[i] at runtime.
// ---------------------------------------------------------------------------
__constant__ int c_opk[NOPS] = {0,3,2,1,3,0,2,3,1,0, 3,2,1,0,3,2,1,3,0,2, 1,3,0,2,1,3,0,1,2,3};
__constant__ int c_opa[NOPS] = {3,1,0,8,4,9,5,0,1,2, 8,7,4,0,5,2,6,3,7,8, 5,2,4,9,3,6,1,0,6,9};
__constant__ int c_opb[NOPS] = {-1,7,-1,-1,2,-1,-1,6,-1,-1, 3,-1,-1,-1,9,-1,-1,1,-1,-1, -1,0,-1,-1,-1,8,-1,-1,-1,5};

// Apply a complex 2x2 gate U on bit position `shift` for all BT elements.
__device__ __forceinline__ void apply_1q(float* s_re, float* s_im, int shift,
                                         float u00r, float u00i, float u01r, float u01i,
                                         float u10r, float u10i, float u11r, float u11i,
                                         int tid)
{
    const int mask = 1 << shift;
    const int low  = mask - 1;
    for (int p = tid; p < BT * (NSTATE / 2); p += TPB) {
        int el = p >> 9, q = p & 511;
        int i0   = ((q >> shift) << (shift + 1)) | (q & low);
        int idx0 = (el << 10) + i0;
        int idx1 = idx0 + mask;
        float a0r = s_re[idx0], a0i = s_im[idx0];
        float a1r = s_re[idx1], a1i = s_im[idx1];
        float n0r = u00r*a0r - u00i*a0i + u01r*a1r - u01i*a1i;
        float n0i = u00r*a0i + u00i*a0r + u01r*a1i + u01i*a1r;
        float n1r = u10r*a0r - u10i*a0i + u11r*a1r - u11i*a1i;
        float n1i = u10r*a0i + u10i*a0r + u11r*a1i + u11i*a1r;
        s_re[idx0] = n0r; s_im[idx0] = n0i;
        s_re[idx1] = n1r; s_im[idx1] = n1i;
    }
}

// CNOT: for control bit = 1, swap target-bit pair.
__device__ __forceinline__ void apply_cnot(float* s_re, float* s_im,
                                           int cshift, int tshift, int tid)
{
    const int b0 = (cshift < tshift) ? cshift : tshift;
    const int b1 = (cshift < tshift) ? tshift : cshift;
    const int cmask = 1 << cshift, tmask = 1 << tshift;
    for (int p = tid; p < BT * (NSTATE / 4); p += TPB) {
        int el = p >> 8, m = p & 255;
        int i = ((m >> b0) << (b0 + 1)) | (m & ((1 << b0) - 1)); // insert 0 at b0
        i     = ((i >> b1) << (b1 + 1)) | (i & ((1 << b1) - 1)); // insert 0 at b1
        int ia = (el << 10) + (i | cmask);   // control=1, target=0
        int ib = ia + tmask;                 // control=1, target=1
        float r = s_re[ia], im = s_im[ia];
        s_re[ia] = s_re[ib]; s_im[ia] = s_im[ib];
        s_re[ib] = r;        s_im[ib] = im;
    }
}

// ---------------------------------------------------------------------------
// Collapse the activation-free MLP chain into (weff[10], beff) -> d_ws.
// out = x @ (W0^T W1^T W2^T W3^T) + (b0 W1^T W2^T W3^T + b1 W2^T W3^T + b2 W3^T + b3)
// ---------------------------------------------------------------------------
__global__ void collapse_mlp_kernel(const float* __restrict__ w0, const float* __restrict__ b0,
                                    const float* __restrict__ w1, const float* __restrict__ b1,
                                    const float* __restrict__ w2, const float* __restrict__ b2,
                                    const float* __restrict__ w3, const float* __restrict__ b3,
                                    float* __restrict__ ws)
{
    if (threadIdx.x != 0 || blockIdx.x != 0) return;
    float r16[16], r32[32], r64[64];
    float beff = b3[0];
    for (int i = 0; i < 16; ++i) { r16[i] = w3[i]; beff += r16[i] * b2[i]; }
    for (int j = 0; j < 32; ++j) {
        float acc = 0.f;
        for (int i = 0; i < 16; ++i) acc += r16[i] * w2[i * 32 + j];
        r32[j] = acc; beff += acc * b1[j];
    }
    for (int k = 0; k < 64; ++k) {
        float acc = 0.f;
        for (int j = 0; j < 32; ++j) acc += r32[j] * w1[j * 64 + k];
        r64[k] = acc; beff += acc * b0[k];
    }
    for (int w = 0; w < NW; ++w) {
        float acc = 0.f;
        for (int k = 0; k < 64; ++k) acc += r64[k] * w0[k * NW + w];
        ws[w] = acc;
    }
    ws[NW] = beff;
}

// ---------------------------------------------------------------------------
// Main kernel: one workgroup simulates BT=16 batch elements with LDS-resident
// statevectors; measurement <Z_w> is a 16x16 f32 WMMA GEMM vs a +-1 sign
// matrix, with K=1024 split across all 8 waves; result fused with the
// collapsed MLP.
// ---------------------------------------------------------------------------
__global__ void __launch_bounds__(TPB)
qsim_kernel(const float* __restrict__ x,    const float* __restrict__ rp,
            const float* __restrict__ rxth, const float* __restrict__ ryth,
            const float* __restrict__ hw,   const float* __restrict__ hb,
            const float* __restrict__ ws,   float* __restrict__ out, int bsz)
{
    extern __shared__ float lds[];
    float* s_re   = lds;                       // BT*1024
    float* s_im   = s_re + BT * NSTATE;        // BT*1024
    float* s_c    = s_im + BT * NSTATE;        // BT*NW  cos(x/2)
    float* s_s    = s_c + BT * NW;             // BT*NW  sin(x/2)
    float* s_feat = s_s + BT * NW;             // 16*16  feats (el x wire)
    float* s_part = s_feat + 16 * 16;          // NWAVES * 16*16 partial tiles

    const int tid     = threadIdx.x;
    const int el_base = blockIdx.x * BT;

    // encoder trig per (element, wire)
    if (tid < BT * NW) {
        int el = tid / NW, w = tid % NW;
        int gb = el_base + el;
        float xv = (gb < bsz) ? x[gb * NW + w] : 0.f;
        s_c[tid] = cosf(0.5f * xv);
        s_s[tid] = sinf(0.5f * xv);
    }
    __syncthreads();

    // product state: amp(i) = prod_w (bit_w ? sin : cos) ; wire 0 = MSB of i
    for (int idx = tid; idx < BT * NSTATE; idx += TPB) {
        int el = idx >> 10, i = idx & (NSTATE - 1);
        float amp = 1.f;
#pragma unroll
        for (int w = 0; w < NW; ++w) {
            int bit = (i >> (9 - w)) & 1;
            amp *= bit ? s_s[el * NW + w] : s_c[el * NW + w];
        }
        s_re[idx] = amp;
        s_im[idx] = 0.f;
    }
    __syncthreads();

    // frozen random layer (angles from rand_params)
    for (int g = 0; g < NOPS; ++g) {
        int kind = c_opk[g], a = c_opa[g];
        if (kind == 3) {
            apply_cnot(s_re, s_im, 9 - a, 9 - c_opb[g], tid);
        } else {
            float th = rp[g];
            float c = cosf(0.5f * th), s = sinf(0.5f * th);
            int sh = 9 - a;
            if (kind == 0)      apply_1q(s_re, s_im, sh, c,0.f,  0.f,-s,  0.f,-s,  c,0.f, tid); // RX
            else if (kind == 1) apply_1q(s_re, s_im, sh, c,0.f,  -s,0.f,  s,0.f,   c,0.f, tid); // RY
            else                apply_1q(s_re, s_im, sh, c,-s,   0.f,0.f, 0.f,0.f, c,s,   tid); // RZ
        }
        __syncthreads();
    }

    // shared RX then RY on every wire -> single fused V = RY*RX per wire
    {
        float cx = cosf(0.5f * rxth[0]), sx = sinf(0.5f * rxth[0]);
        float cy = cosf(0.5f * ryth[0]), sy = sinf(0.5f * ryth[0]);
        float v00r =  cy * cx, v00i =  sy * sx;
        float v01r = -sy * cx, v01i = -cy * sx;
        float v10r =  sy * cx, v10i = -cy * sx;
        float v11r =  cy * cx, v11i = -sy * sx;
        for (int w = 0; w < NW; ++w) {
            apply_1q(s_re, s_im, 9 - w, v00r,v00i, v01r,v01i, v10r,v10i, v11r,v11i, tid);
            __syncthreads();
        }
    }

    // |amp|^2 in place (real plane)
    for (int idx = tid; idx < BT * NSTATE; idx += TPB) {
        float r = s_re[idx], i = s_im[idx];
        s_re[idx] = r * r + i * i;
    }
    __syncthreads();

    // Measurement GEMM, K split across all 8 waves:
    //   feats[16el x 16] = probs[16 x 1024] @ Zsign[1024 x 16]
    //   Zsign[i][n] = (n<10) ? 1-2*bit_{9-n}(i) : 0.   V_WMMA_F32_16X16X4_F32.
    {
        const int lane = tid & 31;
        const int wv   = tid >> 5;      // wave id 0..7, each owns 32 K-chunks
        const int half = lane >> 4;     // K sub-half per ISA A/B layout
        const int m    = lane & 15;     // A: row M   /   B: col N
        v8f acc = {0.f, 0.f, 0.f, 0.f, 0.f, 0.f, 0.f, 0.f};
        const int kk0 = wv * (NSTATE / 4 / NWAVES);
        for (int kk = kk0; kk < kk0 + NSTATE / 4 / NWAVES; ++kk) {
            const int k0 = kk * 4 + half * 2;
            v2f a, b;
            a.x = s_re[m * NSTATE + k0];
            a.y = s_re[m * NSTATE + k0 + 1];
            b.x = (m < NW) ? (1.f - 2.f * (float)(( k0      >> (9 - m)) & 1)) : 0.f;
            b.y = (m < NW) ? (1.f - 2.f * (float)(((k0 + 1) >> (9 - m)) & 1)) : 0.f;
            acc = __builtin_amdgcn_wmma_f32_16x16x4_f32(false, a, false, b,
                                                        (short)0, acc, false, false);
        }
#pragma unroll
        for (int j = 0; j < 8; ++j)               // D layout: vgpr j -> M = j + 8*half
            s_part[wv * 256 + (j + 8 * half) * 16 + m] = acc[j];
    }
    __syncthreads();

    // reduce the 8 partial tiles: one thread per (el, wire) cell
    {
        float f = 0.f;
#pragma unroll
        for (int wv = 0; wv < NWAVES; ++wv) f += s_part[wv * 256 + tid];
        s_feat[tid] = f;
    }
    __syncthreads();

    // q = feats @ head_w + head_b ; fused with collapsed MLP x@weff + beff
    if (tid < BT) {
        int el = tid, gb = el_base + el;
        if (gb < bsz) {
            float q = hb[0];
#pragma unroll
            for (int w = 0; w < NW; ++w) q += s_feat[el * 16 + w] * hw[w];
            float mlp = ws[NW];
#pragma unroll
            for (int w = 0; w < NW; ++w) mlp += x[gb * NW + w] * ws[w];
            out[gb] = q + mlp;
        }
    }
}

extern "C" void kernel_launch(void* const* d_in, const int* in_sizes, int n_in,
                              void* d_out, int out_size, void* d_ws, size_t ws_size,
                              hipStream_t stream)
{
    const float* x    = (const float*)d_in[0];
    const float* rp   = (const float*)d_in[1];
    const float* rxth = (const float*)d_in[2];
    const float* ryth = (const float*)d_in[3];
    const float* hw   = (const float*)d_in[4];
    const float* hb   = (const float*)d_in[5];
    const float* w0   = (const float*)d_in[6];
    const float* b0   = (const float*)d_in[7];
    const float* w1   = (const float*)d_in[8];
    const float* b1   = (const float*)d_in[9];
    const float* w2   = (const float*)d_in[10];
    const float* b2   = (const float*)d_in[11];
    const float* w3   = (const float*)d_in[12];
    const float* b3   = (const float*)d_in[13];
    float* out = (float*)d_out;
    float* ws  = (float*)d_ws;

    const int bsz = in_sizes[0] / NW;

    collapse_mlp_kernel<<<1, 64, 0, stream>>>(w0, b0, w1, b1, w2, b2, w3, b3, ws);

    const size_t lds_bytes =
        (size_t)(2 * BT * NSTATE + 2 * BT * NW + 16 * 16 + NWAVES * 16 * 16)
        * sizeof(float); // ~138 KB
    (void)hipFuncSetAttribute((const void*)qsim_kernel,
                              hipFuncAttributeMaxDynamicSharedMemorySize,
                              (int)lds_bytes);

    const int blocks = (bsz + BT - 1) / BT;
    qsim_kernel<<<blocks, TPB, lds_bytes, stream>>>(x, rp, rxth, ryth, hw, hb, ws, out, bsz);
}